// ACMIL_39754217292330
// MI455X (gfx1250) — compile-verified
//
#include <hip/hip_runtime.h>

// ---------------------------------------------------------------------------
// ACMIL attention pooling for MI455X (gfx1250, wave32).
// Memory-bound: features (204.8MB) read exactly twice (~18us @ 23.3TB/s).
// Scores pass uses V_WMMA_F32_16X16X4_F32 (full f32 precision).
// 7 kernels, zero atomics: all reductions are online-softmax partials merged
// in a fixed order -> bitwise deterministic across replays.
// ---------------------------------------------------------------------------

typedef float v2f __attribute__((ext_vector_type(2)));
typedef float v8f __attribute__((ext_vector_type(8)));

#define N_INST 50000
#define DIM    1024
#define NBR    4
#define TOPK   10
#define RB     120          // stats grid blocks
#define SL     64           // N-slices for bag partials
#define RPS    782          // rows per slice = ceil(50000/64)

// Online softmax helpers: maintain running (max m, sum s of exp(x - m)).
__device__ __forceinline__ void online_ms(float& m, float& s, float x) {
  if (x > m) {                     // first element: m=-inf -> s*exp(-inf)=0
    s = s * expf(m - x) + 1.0f;
    m = x;
  } else {
    s += expf(x - m);
  }
}
__device__ __forceinline__ void merge_ms(float& m1, float& s1, float m2, float s2) {
  const float M = fmaxf(m1, m2);   // both finite when merging valid partials
  s1 = s1 * expf(m1 - M) + s2 * expf(m2 - M);
  m1 = M;
}

// ---------------------------------------------------------------------------
// scores[n,br] = F[n,:] . W[br,:] + b[br]  via V_WMMA_F32_16X16X4_F32.
// One wave per 16-row tile. A layout (16x4 f32): lanes 0-15 v0=K0 v1=K1,
// lanes 16-31 v0=K2 v1=K3. B (4x16): v0 = rows K0/K2, v1 = rows K1/K3,
// column = lane&15; columns >= NBR masked to zero (EXEC stays all-1s).
// ---------------------------------------------------------------------------
__global__ __launch_bounds__(32) void scores_wmma_k(
    const float* __restrict__ F, const float* __restrict__ W,
    const float* __restrict__ Bv, float* __restrict__ S) {
  const int lane = threadIdx.x;
  const int m    = lane & 15;
  const int hi   = lane >> 4;                       // 0 or 1 -> K offset 0 or 2
  const long row = (long)blockIdx.x * 16 + m;
  const float* arow = F + row * DIM + hi * 2;
  const float* brow = W + (long)((m < NBR) ? m : 0) * DIM + hi * 2;
  const float bmask = (m < NBR) ? 1.0f : 0.0f;

  v8f c0 = {0.f,0.f,0.f,0.f,0.f,0.f,0.f,0.f};
  v8f c1 = {0.f,0.f,0.f,0.f,0.f,0.f,0.f,0.f};
  for (int k = 0; k < DIM; k += 8) {                // 2 WMMAs / iter for ILP
    v2f a0, b0, a1, b1;
    a0.x = arow[k];     a0.y = arow[k + 1];
    b0.x = brow[k] * bmask;     b0.y = brow[k + 1] * bmask;
    a1.x = arow[k + 4]; a1.y = arow[k + 5];
    b1.x = brow[k + 4] * bmask; b1.y = brow[k + 5] * bmask;
    c0 = __builtin_amdgcn_wmma_f32_16x16x4_f32(false, a0, false, b0,
                                               (short)0, c0, false, false);
    c1 = __builtin_amdgcn_wmma_f32_16x16x4_f32(false, a1, false, b1,
                                               (short)0, c1, false, false);
  }
  // C/D layout: VGPR i holds M=i (lanes 0-15) and M=8+i (lanes 16-31), N=lane&15.
  if (m < NBR) {
    const float bias = Bv[m];
    const long mbase = (long)blockIdx.x * 16 + hi * 8;
    S[(mbase + 0) * NBR + m] = c0[0] + c1[0] + bias;
    S[(mbase + 1) * NBR + m] = c0[1] + c1[1] + bias;
    S[(mbase + 2) * NBR + m] = c0[2] + c1[2] + bias;
    S[(mbase + 3) * NBR + m] = c0[3] + c1[3] + bias;
    S[(mbase + 4) * NBR + m] = c0[4] + c1[4] + bias;
    S[(mbase + 5) * NBR + m] = c0[5] + c1[5] + bias;
    S[(mbase + 6) * NBR + m] = c0[6] + c1[6] + bias;
    S[(mbase + 7) * NBR + m] = c0[7] + c1[7] + bias;
  }
}

// Per-branch partial (max, sum-exp) over N in ONE pass (online softmax).
// 120*256 = 30720 threads < N, so every thread/partial is valid (finite max).
__global__ __launch_bounds__(256) void bstats_k(const float* __restrict__ S,
                                                float* __restrict__ pmax,   // [RB][4]
                                                float* __restrict__ psum) { // [RB][4]
  float m[4] = {-__builtin_inff(), -__builtin_inff(),
                -__builtin_inff(), -__builtin_inff()};
  float s[4] = {0.f, 0.f, 0.f, 0.f};
  for (int n = blockIdx.x * 256 + threadIdx.x; n < N_INST; n += RB * 256) {
    const float4 v = reinterpret_cast<const float4*>(S)[n];
    online_ms(m[0], s[0], v.x); online_ms(m[1], s[1], v.y);
    online_ms(m[2], s[2], v.z); online_ms(m[3], s[3], v.w);
  }
  __shared__ float sm[256], ss[256];
  for (int br = 0; br < 4; ++br) {
    sm[threadIdx.x] = m[br];
    ss[threadIdx.x] = s[br];
    __syncthreads();
    for (int st = 128; st > 0; st >>= 1) {
      if (threadIdx.x < st)
        merge_ms(sm[threadIdx.x], ss[threadIdx.x],
                 sm[threadIdx.x + st], ss[threadIdx.x + st]);
      __syncthreads();
    }
    if (threadIdx.x == 0) {
      pmax[blockIdx.x * 4 + br] = sm[0];
      psum[blockIdx.x * 4 + br] = ss[0];
    }
    __syncthreads();
  }
}

// w[n] = mean_br softmax_n(scores). Each block serially merges the 120
// partials in fixed order (identical result in every block -> deterministic).
__global__ __launch_bounds__(256) void compute_w_k(
    const float* __restrict__ S, const float* __restrict__ pmax,
    const float* __restrict__ psum, float* __restrict__ w) {
  __shared__ float gm[4], ginv[4];
  if (threadIdx.x < NBR) {
    float M = pmax[threadIdx.x];
    float Sm = psum[threadIdx.x];
    for (int i = 1; i < RB; ++i)
      merge_ms(M, Sm, pmax[i * 4 + threadIdx.x], psum[i * 4 + threadIdx.x]);
    gm[threadIdx.x]   = M;
    ginv[threadIdx.x] = 1.0f / Sm;
  }
  __syncthreads();
  const int n = blockIdx.x * 256 + threadIdx.x;
  if (n >= N_INST) return;
  const float4 v = reinterpret_cast<const float4*>(S)[n];
  w[n] = 0.25f * (expf(v.x - gm[0]) * ginv[0] + expf(v.y - gm[1]) * ginv[1] +
                  expf(v.z - gm[2]) * ginv[2] + expf(v.w - gm[3]) * ginv[3]);
}

// Single-block top-10 select & zero. w > 0, so float bits are order-preserving;
// pack (bits<<32)|idx -> unique keys, per-thread sorted top-10, then 10 rounds
// of block max. Fully deterministic.
__global__ __launch_bounds__(1024) void topk_zero_k(float* __restrict__ w) {
  const int tid = threadIdx.x;
  unsigned long long loc[TOPK];
#pragma unroll
  for (int i = 0; i < TOPK; ++i) loc[i] = 0ull;
  for (int n = tid; n < N_INST; n += 1024) {
    const unsigned long long key =
        ((unsigned long long)__float_as_uint(w[n]) << 32) | (unsigned int)n;
    if (key > loc[TOPK - 1]) {
      int j = TOPK - 1;
      while (j > 0 && loc[j - 1] < key) { loc[j] = loc[j - 1]; --j; }
      loc[j] = key;
    }
  }
  __shared__ unsigned long long red[1024];
  int ptr = 0;
  for (int it = 0; it < TOPK; ++it) {
    const unsigned long long cand = (ptr < TOPK) ? loc[ptr] : 0ull;
    red[tid] = cand;
    __syncthreads();
    for (int st = 512; st > 0; st >>= 1) {
      if (tid < st) {
        const unsigned long long o = red[tid + st];
        if (o > red[tid]) red[tid] = o;
      }
      __syncthreads();
    }
    const unsigned long long best = red[0];
    __syncthreads();
    if (cand == best && best != 0ull) ++ptr;     // winner consumes its entry
    if (tid == 0 && best != 0ull)
      w[(int)(best & 0xffffffffull)] = 0.0f;
  }
}

// Partial (max, sum-exp) of (zeroed) w in one pass.
__global__ __launch_bounds__(256) void wstats_k(const float* __restrict__ w,
                                                float* __restrict__ pmax2,  // [RB]
                                                float* __restrict__ psum2) {// [RB]
  float m = -__builtin_inff(), s = 0.f;
  for (int n = blockIdx.x * 256 + threadIdx.x; n < N_INST; n += RB * 256)
    online_ms(m, s, w[n]);
  __shared__ float sm[256], ss[256];
  sm[threadIdx.x] = m;
  ss[threadIdx.x] = s;
  __syncthreads();
  for (int st = 128; st > 0; st >>= 1) {
    if (threadIdx.x < st)
      merge_ms(sm[threadIdx.x], ss[threadIdx.x],
               sm[threadIdx.x + st], ss[threadIdx.x + st]);
    __syncthreads();
  }
  if (threadIdx.x == 0) {
    pmax2[blockIdx.x] = sm[0];
    psum2[blockIdx.x] = ss[0];
  }
}

// bag partials + fused w finalization. Block (cx, slice): merge the 120
// (m,s) partials serially (identical in all blocks), compute final
// wn = exp(w-M)/S for this slice's rows into LDS (hot loop stays
// ds_broadcast + global_load + fmaf), write final w to d_out from the
// cx==0 column-blocks, accumulate 256 contiguous columns over ~782 rows.
__global__ __launch_bounds__(256) void bag_partial_k(
    const float* __restrict__ F, const float* __restrict__ w,
    const float* __restrict__ pmax2, const float* __restrict__ psum2,
    float* __restrict__ bagp, float* __restrict__ wout) {
  __shared__ float sM, sInv;
  __shared__ float wn_s[RPS];
  const int tid = threadIdx.x;
  if (tid == 0) {
    float M = pmax2[0], Sm = psum2[0];
    for (int i = 1; i < RB; ++i) merge_ms(M, Sm, pmax2[i], psum2[i]);
    sM = M;
    sInv = 1.0f / Sm;
  }
  __syncthreads();
  const int slice = blockIdx.y;
  const int r0 = slice * RPS;
  int rows = N_INST - r0;
  if (rows > RPS) rows = RPS;
  const float M = sM, inv = sInv;
  for (int r = tid; r < rows; r += 256) {
    const float wn = expf(w[r0 + r] - M) * inv;
    wn_s[r] = wn;
    if (blockIdx.x == 0) wout[r0 + r] = wn;   // final w output, race-free
  }
  __syncthreads();
  const int col = blockIdx.x * 256 + tid;
  float acc = 0.f;
  for (int r = 0; r < rows; ++r)
    acc = fmaf(wn_s[r], F[(long)(r0 + r) * DIM + col], acc);
  bagp[(long)slice * DIM + col] = acc;
}

// Deterministic fixed-order combine over slices.
__global__ __launch_bounds__(256) void bag_combine_k(const float* __restrict__ bagp,
                                                     float* __restrict__ bag) {
  const int col = blockIdx.x * 256 + threadIdx.x;
  float acc = 0.f;
  for (int s = 0; s < SL; ++s) acc += bagp[(long)s * DIM + col];
  bag[col] = acc;
}

extern "C" void kernel_launch(void* const* d_in, const int* in_sizes, int n_in,
                              void* d_out, int out_size, void* d_ws, size_t ws_size,
                              hipStream_t stream) {
  const float* F  = (const float*)d_in[0];   // [50000,1024]
  const float* W  = (const float*)d_in[1];   // [4,1024]
  const float* Bv = (const float*)d_in[2];   // [4]
  float* out  = (float*)d_out;
  float* bag  = out;                         // [1024]
  float* wout = out + DIM;                   // [50000] final weights

  float* ws    = (float*)d_ws;               // ~1.27 MB used
  float* S     = ws;                         // scores        [200000]
  float* WBUF  = ws + 200000;                // intermediate w [50000]
  float* PMAX  = ws + 250000;                // [RB*4]
  float* PSUM  = ws + 250480;                // [RB*4]
  float* PMAX2 = ws + 250960;                // [RB]
  float* PSUM2 = ws + 251080;                // [RB]
  float* BAGP  = ws + 251200;                // [SL*DIM]

  scores_wmma_k<<<N_INST / 16, 32, 0, stream>>>(F, W, Bv, S);
  bstats_k<<<RB, 256, 0, stream>>>(S, PMAX, PSUM);
  compute_w_k<<<(N_INST + 255) / 256, 256, 0, stream>>>(S, PMAX, PSUM, WBUF);
  topk_zero_k<<<1, 1024, 0, stream>>>(WBUF);
  wstats_k<<<RB, 256, 0, stream>>>(WBUF, PMAX2, PSUM2);
  bag_partial_k<<<dim3(DIM / 256, SL), 256, 0, stream>>>(F, WBUF, PMAX2, PSUM2,
                                                         BAGP, wout);
  bag_combine_k<<<DIM / 256, 256, 0, stream>>>(BAGP, bag);
}